// GNNLinkPredict_33947421508070
// MI455X (gfx1250) — compile-verified
//
#include <hip/hip_runtime.h>

typedef __attribute__((ext_vector_type(2))) float v2f;
typedef __attribute__((ext_vector_type(8))) float v8f;

__device__ __forceinline__ v8f wmma_f32_k4(v2f a, v2f b, v8f c) {
  // V_WMMA_F32_16X16X4_F32 : D = A(16x4) * B(4x16) + C, full fp32
  return __builtin_amdgcn_wmma_f32_16x16x4_f32(false, a, false, b, (short)0, c,
                                               false, false);
}

// Monotonic float atomic-max trick (works with -inf init).
__device__ __forceinline__ void atomicMaxF(float* addr, float val) {
  if (val >= 0.0f)
    atomicMax((int*)addr, __float_as_int(val));
  else
    atomicMin((unsigned int*)addr, __float_as_uint(val));
}

// ---------------------------------------------------------------------------
// GEMM + fused attention coefficients:
//   h[n x 128] = in[n x 128] @ W[128 x 128]           (fp32 WMMA 16x16x4)
//   a_s[n,head] += sum_col h[n,col]*att_src[col]       (head = col>>6)
//   a_d[n,head] += sum_col h[n,col]*att_dst[col]
// Block: 256 thr = 8 waves; block covers 16 rows x 128 cols, wave w -> cols 16w..
// a_s/a_d must be zeroed before launch (k_init_att).
// ---------------------------------------------------------------------------
__global__ void k_gemm128_att(const float* __restrict__ A,
                              const float* __restrict__ W,
                              float* __restrict__ out,
                              const float* __restrict__ att_src,
                              const float* __restrict__ att_dst,
                              float* __restrict__ a_s, float* __restrict__ a_d,
                              int nrows) {
  const int lane = threadIdx.x & 31;
  const int wave = threadIdx.x >> 5;
  const int m0 = blockIdx.x * 16;
  const int n0 = wave * 16;
  const int hl = lane >> 4;             // 0: lanes 0-15, 1: lanes 16-31
  const int ll = lane & 15;
  int row = m0 + ll;
  if (row >= nrows) row = nrows - 1;    // clamp loads; stores guarded below
  const int col = n0 + ll;
  const int khalf = hl * 2;             // K offset 0 or 2 within the 4-step
  const float* arow = A + (size_t)row * 128;

  v8f acc = {};
  for (int k0 = 0; k0 < 128; k0 += 4) {
    const int k = k0 + khalf;
    v2f a, b;
    a.x = arow[k];
    a.y = arow[k + 1];
    b.x = W[(size_t)k * 128 + col];
    b.y = W[(size_t)(k + 1) * 128 + col];
    acc = wmma_f32_k4(a, b, acc);
  }

  // ---- store D tile (fast path for full tiles) ----
  const int rbase = m0 + hl * 8;
  if (m0 + 16 <= nrows) {
#pragma unroll
    for (int r = 0; r < 8; ++r)
      out[(size_t)(rbase + r) * 128 + col] = acc[r];
  } else {
#pragma unroll
    for (int r = 0; r < 8; ++r)
      if (rbase + r < nrows) out[(size_t)(rbase + r) * 128 + col] = acc[r];
  }

  // ---- fused attention coefficients ----
  const float wsa = att_src[col];       // att flattened [H*C]=[128]
  const float wda = att_dst[col];
  float ps[8], pd[8];
#pragma unroll
  for (int r = 0; r < 8; ++r) { ps[r] = acc[r] * wsa; pd[r] = acc[r] * wda; }
#pragma unroll
  for (int mk = 8; mk >= 1; mk >>= 1) { // reduce the 16 cols of each half-wave
#pragma unroll
    for (int r = 0; r < 8; ++r) {
      ps[r] += __shfl_xor(ps[r], mk, 32);
      pd[r] += __shfl_xor(pd[r], mk, 32);
    }
  }
  if (ll == 0) {                        // lanes 0 and 16 hold the row sums
    const int head = n0 >> 6;           // waves 0-3 -> head 0, 4-7 -> head 1
#pragma unroll
    for (int r = 0; r < 8; ++r) {
      const int orow = rbase + r;
      if (orow < nrows) {
        atomicAdd(&a_s[orow * 2 + head], ps[r]);
        atomicAdd(&a_d[orow * 2 + head], pd[r]);
      }
    }
  }
}

// ---------------------------------------------------------------------------
// Init kernels.
// ---------------------------------------------------------------------------
__global__ void k_init_att(float* __restrict__ a_s, float* __restrict__ a_d,
                           int n) {
  const int i = blockIdx.x * blockDim.x + threadIdx.x;
  if (i < n * 2) { a_s[i] = 0.f; a_d[i] = 0.f; }
}

__global__ void k_init_seg(float* __restrict__ m, float* __restrict__ den,
                           float* __restrict__ out, int n) {
  const int tid = blockIdx.x * blockDim.x + threadIdx.x;
  const int stride = gridDim.x * blockDim.x;
  const int total = n * 128;
  for (int i = tid; i < total; i += stride) out[i] = 0.f;
  for (int i = tid; i < n * 2; i += stride) {
    m[i] = -__builtin_inff();
    den[i] = 0.f;
  }
}

__device__ __forceinline__ void edge_nodes(const long long* __restrict__ src,
                                           const long long* __restrict__ dst,
                                           int e, int nE, int& s, int& d) {
  if (e < nE) { s = (int)src[e]; d = (int)dst[e]; }
  else        { s = e - nE; d = s; }       // appended self-loops
}

// Pass 1: segment max of leaky-relu scores. Thread per edge.
__global__ void k_edge_max(const float* __restrict__ a_s,
                           const float* __restrict__ a_d,
                           const long long* __restrict__ src,
                           const long long* __restrict__ dst,
                           float* __restrict__ m, int nE, int nN) {
  const int e = blockIdx.x * blockDim.x + threadIdx.x;
  if (e >= nE + nN) return;
  int s, d;
  edge_nodes(src, dst, e, nE, s, d);
#pragma unroll
  for (int h = 0; h < 2; ++h) {
    float sc = a_s[s * 2 + h] + a_d[d * 2 + h];
    sc = sc > 0.f ? sc : 0.2f * sc;
    atomicMaxF(&m[d * 2 + h], sc);
  }
}

// Pass 2: denom[d,h] += exp(score - m[d,h]). Thread per edge.
__global__ void k_edge_expsum(const float* __restrict__ a_s,
                              const float* __restrict__ a_d,
                              const long long* __restrict__ src,
                              const long long* __restrict__ dst,
                              const float* __restrict__ m,
                              float* __restrict__ den, int nE, int nN) {
  const int e = blockIdx.x * blockDim.x + threadIdx.x;
  if (e >= nE + nN) return;
  int s, d;
  edge_nodes(src, dst, e, nE, s, d);
#pragma unroll
  for (int h = 0; h < 2; ++h) {
    float sc = a_s[s * 2 + h] + a_d[d * 2 + h];
    sc = sc > 0.f ? sc : 0.2f * sc;
    atomicAdd(&den[d * 2 + h], __expf(sc - m[d * 2 + h]));
  }
}

// Pass 3: out[d] += alpha * h[src]. One wave per edge, 4 channels per lane.
__global__ void k_edge_agg(const float* __restrict__ hfeat,
                           const float* __restrict__ a_s,
                           const float* __restrict__ a_d,
                           const float* __restrict__ m,
                           const float* __restrict__ den,
                           const long long* __restrict__ src,
                           const long long* __restrict__ dst,
                           float* __restrict__ out, int nE, int nN) {
  const int lane = threadIdx.x & 31;
  const int e = (blockIdx.x * blockDim.x + threadIdx.x) >> 5;
  if (e >= nE + nN) return;
  int s, d;
  edge_nodes(src, dst, e, nE, s, d);
  const int head = lane >> 4;            // channel = lane*4+i, head = channel>>6
  float sc = a_s[s * 2 + head] + a_d[d * 2 + head];
  sc = sc > 0.f ? sc : 0.2f * sc;
  const float alpha = __expf(sc - m[d * 2 + head]) / den[d * 2 + head];
  const float* hs = hfeat + (size_t)s * 128;
  float* od = out + (size_t)d * 128;
  const int c0 = lane * 4;
#pragma unroll
  for (int i = 0; i < 4; ++i) atomicAdd(&od[c0 + i], hs[c0 + i] * alpha);
}

__global__ void k_bias_act(float* __restrict__ x, const float* __restrict__ bias,
                           int n, int do_relu) {
  const int i = blockIdx.x * blockDim.x + threadIdx.x;
  if (i >= n * 128) return;
  float v = x[i] + bias[i & 127];
  if (do_relu) v = v > 0.f ? v : 0.f;
  x[i] = v;
}

// ---------------------------------------------------------------------------
// Decoder: out[l] = relu(concat(z[ei0],z[ei1]) @ lin1_w + b1) @ lin2_w + b2
// One wave per 16 label edges; 16x256 @ 256x64 via fp32 WMMA, fused epilogue.
// lin1_w is staged once per block into LDS, transposed to [col][k] with
// stride 260 floats: bank-conflict-free for the B-fragment pattern and lets
// each lane fetch its (k, k+1) pair as a single ds_load_b64.
// Dynamic LDS: 64 * 260 * 4 = 66560 bytes (4 blocks per 320KB WGP).
// ---------------------------------------------------------------------------
#define DEC_WSTRIDE 260

__global__ void k_decoder(const float* __restrict__ z,
                          const long long* __restrict__ eli,  // [2, L]
                          const float* __restrict__ l1w,      // [256, 64]
                          const float* __restrict__ l1b,      // [64]
                          const float* __restrict__ l2w,      // [64]
                          const float* __restrict__ l2b,      // [1]
                          float* __restrict__ out, int L) {
  extern __shared__ float wlds[];       // [64][DEC_WSTRIDE]

  // Cooperative transpose-stage of lin1_w (coalesced global reads).
  for (int idx = threadIdx.x; idx < 256 * 64; idx += blockDim.x) {
    const int k = idx >> 6;
    const int c = idx & 63;
    wlds[c * DEC_WSTRIDE + k] = l1w[idx];
  }
  __syncthreads();

  const int lane = threadIdx.x & 31;
  const int wid = blockIdx.x * (blockDim.x >> 5) + (threadIdx.x >> 5);
  const int e0 = wid * 16;
  if (e0 >= L) return;                  // after the barrier: block-safe
  const int ll = lane & 15;
  const int hl = lane >> 4;
  int eidx = e0 + ll;
  if (eidx >= L) eidx = L - 1;          // clamp loads; stores guarded
  const float* ps = z + (size_t)((int)eli[eidx]) * 128;              // k 0..127
  const float* pd = z + (size_t)((int)eli[(size_t)L + eidx]) * 128;  // 128..255
  const int khalf = hl * 2;

  const float* w0r = wlds + (size_t)ll * DEC_WSTRIDE;
  const float* w1r = w0r + 16 * DEC_WSTRIDE;
  const float* w2r = w0r + 32 * DEC_WSTRIDE;
  const float* w3r = w0r + 48 * DEC_WSTRIDE;

  v8f acc0 = {}, acc1 = {}, acc2 = {}, acc3 = {};
  for (int k0 = 0; k0 < 256; k0 += 4) {
    const int k = k0 + khalf;           // k is even -> float2 aligned
    v2f a;
    if (k0 < 128) { a.x = ps[k];       a.y = ps[k + 1]; }
    else          { a.x = pd[k - 128]; a.y = pd[k - 127]; }
    const float2 t0 = *(const float2*)(w0r + k);
    const float2 t1 = *(const float2*)(w1r + k);
    const float2 t2 = *(const float2*)(w2r + k);
    const float2 t3 = *(const float2*)(w3r + k);
    v2f b0, b1, b2, b3;
    b0.x = t0.x; b0.y = t0.y;
    b1.x = t1.x; b1.y = t1.y;
    b2.x = t2.x; b2.y = t2.y;
    b3.x = t3.x; b3.y = t3.y;
    acc0 = wmma_f32_k4(a, b0, acc0);
    acc1 = wmma_f32_k4(a, b1, acc1);
    acc2 = wmma_f32_k4(a, b2, acc2);
    acc3 = wmma_f32_k4(a, b3, acc3);
  }

  // Epilogue: bias + relu + dot with lin2_w, per-row accumulation.
  float racc[8];
#pragma unroll
  for (int r = 0; r < 8; ++r) racc[r] = 0.f;
  {
    const int c = ll;
    const float bb = l1b[c], w2 = l2w[c];
#pragma unroll
    for (int r = 0; r < 8; ++r) { float v = acc0[r] + bb; racc[r] += (v > 0.f ? v : 0.f) * w2; }
  }
  {
    const int c = ll + 16;
    const float bb = l1b[c], w2 = l2w[c];
#pragma unroll
    for (int r = 0; r < 8; ++r) { float v = acc1[r] + bb; racc[r] += (v > 0.f ? v : 0.f) * w2; }
  }
  {
    const int c = ll + 32;
    const float bb = l1b[c], w2 = l2w[c];
#pragma unroll
    for (int r = 0; r < 8; ++r) { float v = acc2[r] + bb; racc[r] += (v > 0.f ? v : 0.f) * w2; }
  }
  {
    const int c = ll + 48;
    const float bb = l1b[c], w2 = l2w[c];
#pragma unroll
    for (int r = 0; r < 8; ++r) { float v = acc3[r] + bb; racc[r] += (v > 0.f ? v : 0.f) * w2; }
  }
  // Reduce over the 16 lanes of each half-wave (rows r / rows r+8).
#pragma unroll
  for (int mk = 8; mk >= 1; mk >>= 1) {
#pragma unroll
    for (int r = 0; r < 8; ++r) racc[r] += __shfl_xor(racc[r], mk, 32);
  }
  if (ll == 0) {
    const float bb2 = l2b[0];
    const int rb = e0 + hl * 8;
#pragma unroll
    for (int r = 0; r < 8; ++r)
      if (rb + r < L) out[rb + r] = racc[r] + bb2;
  }
}

// ---------------------------------------------------------------------------
extern "C" void kernel_launch(void* const* d_in, const int* in_sizes, int n_in,
                              void* d_out, int out_size, void* d_ws, size_t ws_size,
                              hipStream_t stream) {
  const float*     x    = (const float*)d_in[0];
  const long long* ei   = (const long long*)d_in[1];   // int64 [2,E]
  const long long* eli  = (const long long*)d_in[2];   // int64 [2,L]
  const float*     W1   = (const float*)d_in[3];
  const float*     as1  = (const float*)d_in[4];
  const float*     ad1  = (const float*)d_in[5];
  const float*     b1   = (const float*)d_in[6];
  const float*     W2   = (const float*)d_in[7];
  const float*     as2  = (const float*)d_in[8];
  const float*     ad2  = (const float*)d_in[9];
  const float*     b2   = (const float*)d_in[10];
  const float*     l1w  = (const float*)d_in[11];
  const float*     l1b  = (const float*)d_in[12];
  const float*     l2w  = (const float*)d_in[13];
  const float*     l2b  = (const float*)d_in[14];
  float* out = (float*)d_out;

  const int n = in_sizes[0] / 128;   // 100000
  const int e = in_sizes[1] / 2;     // 1600000
  const int l = in_sizes[2] / 2;     // 500000
  const int totE = e + n;

  float* ws   = (float*)d_ws;
  float* bufA = ws;                           // [n,128] transformed features h
  float* bufB = bufA + (size_t)n * 128;       // [n,128] aggregation target
  float* a_s  = bufB + (size_t)n * 128;       // [n,2]
  float* a_d  = a_s + (size_t)n * 2;          // [n,2]
  float* mseg = a_d + (size_t)n * 2;          // [n,2] segment max
  float* den  = mseg + (size_t)n * 2;         // [n,2] segment denom

  const long long* src = ei;
  const long long* dst = ei + e;

  const int gGemm = (n + 15) / 16;
  const int gAtt  = (n * 2 + 255) / 256;
  const int gEdge = (totE + 255) / 256;
  const int gWaveE = (totE + 7) / 8;          // wave-per-edge
  const int gElem = (n * 128 + 255) / 256;

  // ---------------- Layer 1 ----------------
  k_init_att<<<gAtt, 256, 0, stream>>>(a_s, a_d, n);
  k_gemm128_att<<<gGemm, 256, 0, stream>>>(x, W1, bufA, as1, ad1, a_s, a_d, n);
  k_init_seg<<<2048, 256, 0, stream>>>(mseg, den, bufB, n);
  k_edge_max<<<gEdge, 256, 0, stream>>>(a_s, a_d, src, dst, mseg, e, n);
  k_edge_expsum<<<gEdge, 256, 0, stream>>>(a_s, a_d, src, dst, mseg, den, e, n);
  k_edge_agg<<<gWaveE, 256, 0, stream>>>(bufA, a_s, a_d, mseg, den, src, dst,
                                         bufB, e, n);
  k_bias_act<<<gElem, 256, 0, stream>>>(bufB, b1, n, 1);   // h1 = relu(o1+b1)

  // ---------------- Layer 2 ----------------
  k_init_att<<<gAtt, 256, 0, stream>>>(a_s, a_d, n);
  k_gemm128_att<<<gGemm, 256, 0, stream>>>(bufB, W2, bufA, as2, ad2, a_s, a_d, n);
  k_init_seg<<<2048, 256, 0, stream>>>(mseg, den, bufB, n);  // bufB now free
  k_edge_max<<<gEdge, 256, 0, stream>>>(a_s, a_d, src, dst, mseg, e, n);
  k_edge_expsum<<<gEdge, 256, 0, stream>>>(a_s, a_d, src, dst, mseg, den, e, n);
  k_edge_agg<<<gWaveE, 256, 0, stream>>>(bufA, a_s, a_d, mseg, den, src, dst,
                                         bufB, e, n);
  k_bias_act<<<gElem, 256, 0, stream>>>(bufB, b2, n, 0);   // z = o2 + b2

  // ---------------- Decoder ----------------
  const int decWaves = (l + 15) / 16;
  const size_t decLds = (size_t)64 * DEC_WSTRIDE * sizeof(float);
  k_decoder<<<(decWaves + 7) / 8, 256, decLds, stream>>>(bufB, eli, l1w, l1b,
                                                         l2w, l2b, out, l);
}